// LocalAttention_56813827391596
// MI455X (gfx1250) — compile-verified
//
#include <hip/hip_runtime.h>

// Problem constants (match the reference).
#define B_    2
#define H_    12
#define L_    2048
#define D_    64
#define HALF_ 64          // WINDOW/2
#define SCALE_ 0.125f     // 1/sqrt(64)

#define NT_ 9             // column tiles covering [i0-64, i0+80) for a 16-row tile
#define WAVES_PER_BLOCK 4

typedef float v2f __attribute__((ext_vector_type(2)));
typedef float v4f __attribute__((ext_vector_type(4)));   // clang vector: legal for nontemporal builtins
typedef float v8f __attribute__((ext_vector_type(8)));

// ---------------------------------------------------------------------------
// Kernel 1: zero the attn region (B*H*L*L floats = 402 MB) with 128-bit
// NONTEMPORAL stores. Working set is ~2x the 192MB L2 with zero reuse, so
// TH=NT (store policy 1) avoids thrashing L2 for the WMMA kernel's K/V reads.
// ---------------------------------------------------------------------------
__global__ void attn_zero_kernel(v4f* __restrict__ p, long long n4) {
    long long i      = (long long)blockIdx.x * blockDim.x + threadIdx.x;
    long long stride = (long long)gridDim.x * blockDim.x;
    const v4f z = {0.f, 0.f, 0.f, 0.f};
    for (; i < n4; i += stride)
        __builtin_nontemporal_store(z, p + i);
}

// ---------------------------------------------------------------------------
// Kernel 2: sliding-window attention. One wave32 per 16-row tile.
//
// Layout assumptions (CDNA5 ISA 7.12.2, wave32):
//   A (16x4 f32, MxK):  lane = M + 16*(Ksub>=2), vgpr = Ksub%2
//   B (4x16 f32, KxN):  lane = N + 16*(K>=2),    vgpr = K%2
//   C/D (16x16 f32):    lane = N + 16*(M>=8),    vgpr = M%8
// ---------------------------------------------------------------------------
__launch_bounds__(32 * WAVES_PER_BLOCK)
__global__ void local_attn_kernel(const float* __restrict__ Q,
                                  const float* __restrict__ K,
                                  const float* __restrict__ V,
                                  float* __restrict__ out,
                                  float* __restrict__ attn) {
    // Per-wave LDS pad for the P (C-layout) -> A-layout transpose bounce.
    // Stride 18 floats: even (keeps ds_load_b64 8B-aligned), avoids conflicts.
    __shared__ float ldsP[WAVES_PER_BLOCK][16 * 18];

    const int lane = threadIdx.x & 31;
    const int wave = threadIdx.x >> 5;
    const int hi   = lane >> 4;     // 0: lanes 0-15, 1: lanes 16-31
    const int ln   = lane & 15;

    const int tile = blockIdx.x * WAVES_PER_BLOCK + wave;   // global 16-row tile
    const int bh   = tile >> 7;                             // / (L/16 = 128)
    const int i0   = (tile & 127) << 4;

    const float* Qb = Q + (size_t)bh * L_ * D_;
    const float* Kb = K + (size_t)bh * L_ * D_;
    const float* Vb = V + (size_t)bh * L_ * D_;
    float*       Ob = out  + (size_t)bh * L_ * D_;
    float*       Ab = attn + (size_t)bh * L_ * L_;

    // ---- Load Q tile as 16 A-fragments (K-dim 64 = 16 sub-K of 4) ----------
    // A frag for sub-k kk: lane holds Q[i0+ln][4*kk + 2*hi + {0,1}]
    v2f aQ[16];
    {
        const float* qrow = Qb + (size_t)(i0 + ln) * D_ + 2 * hi;
        #pragma unroll
        for (int kk = 0; kk < 16; ++kk)
            aQ[kk] = *(const v2f*)(qrow + 4 * kk);
    }

    // ---- S = Q K^T over the 9-tile band ------------------------------------
    v8f S[NT_];
    #pragma unroll
    for (int t = 0; t < NT_; ++t) {
        const int j0 = i0 - HALF_ + 16 * t;
        v8f c = {0.f, 0.f, 0.f, 0.f, 0.f, 0.f, 0.f, 0.f};
        if (j0 >= 0 && j0 < L_) {
            // B frag for sub-k kk: lane holds K[j0+ln][4*kk + 2*hi + {0,1}]
            const float* krow = Kb + (size_t)(j0 + ln) * D_ + 2 * hi;
            #pragma unroll
            for (int kk = 0; kk < 16; ++kk) {
                v2f bK = *(const v2f*)(krow + 4 * kk);
                c = __builtin_amdgcn_wmma_f32_16x16x4_f32(
                        false, aQ[kk], false, bK, (short)0, c, false, false);
            }
        }
        S[t] = c;
    }

    // ---- scale + window mask + row max -------------------------------------
    const float NEGINF = -__builtin_inff();
    float m[8];
    #pragma unroll
    for (int v = 0; v < 8; ++v) m[v] = NEGINF;

    #pragma unroll
    for (int t = 0; t < NT_; ++t) {
        const int j0 = i0 - HALF_ + 16 * t;
        const bool tv = (j0 >= 0 && j0 < L_);
        const int j = j0 + ln;
        #pragma unroll
        for (int v = 0; v < 8; ++v) {
            const int i = i0 + v + 8 * hi;               // C layout: row = v + 8*hi
            const bool ok = tv && (j >= i - HALF_) && (j < i + HALF_);
            const float s = ok ? S[t][v] * SCALE_ : NEGINF;
            S[t][v] = s;
            m[v] = fmaxf(m[v], s);
        }
    }
    // Row elements live in one VGPR across a 16-lane half: xor-reduce 1/2/4/8.
    #pragma unroll
    for (int v = 0; v < 8; ++v) {
        float x = m[v];
        x = fmaxf(x, __shfl_xor(x, 1, 32));
        x = fmaxf(x, __shfl_xor(x, 2, 32));
        x = fmaxf(x, __shfl_xor(x, 4, 32));
        x = fmaxf(x, __shfl_xor(x, 8, 32));
        m[v] = x;
    }

    // ---- exp, row sum, normalize ------------------------------------------
    float sum[8];
    #pragma unroll
    for (int v = 0; v < 8; ++v) sum[v] = 0.f;
    #pragma unroll
    for (int t = 0; t < NT_; ++t) {
        #pragma unroll
        for (int v = 0; v < 8; ++v) {
            const float p = __expf(S[t][v] - m[v]);      // exp(-inf)=0 on masked
            S[t][v] = p;
            sum[v] += p;
        }
    }
    #pragma unroll
    for (int v = 0; v < 8; ++v) {
        float x = sum[v];
        x += __shfl_xor(x, 1, 32);
        x += __shfl_xor(x, 2, 32);
        x += __shfl_xor(x, 4, 32);
        x += __shfl_xor(x, 8, 32);
        sum[v] = 1.f / x;                                 // j=i always valid
    }
    #pragma unroll
    for (int t = 0; t < NT_; ++t)
        #pragma unroll
        for (int v = 0; v < 8; ++v)
            S[t][v] *= sum[v];

    // ---- write attn band + O = P V -----------------------------------------
    v8f O[4];
    #pragma unroll
    for (int dc = 0; dc < 4; ++dc) {
        v8f z = {0.f, 0.f, 0.f, 0.f, 0.f, 0.f, 0.f, 0.f};
        O[dc] = z;
    }

    float* lw = ldsP[wave];

    #pragma unroll
    for (int t = 0; t < NT_; ++t) {
        const int j0 = i0 - HALF_ + 16 * t;
        if (j0 < 0 || j0 >= L_) continue;                 // zero tile: skip
        const int j = j0 + ln;

        // attn store: streaming, write-once, no reuse -> nontemporal.
        #pragma unroll
        for (int v = 0; v < 8; ++v) {
            const int i = i0 + v + 8 * hi;
            __builtin_nontemporal_store(S[t][v], Ab + (size_t)i * L_ + j);
        }

        // C-layout -> LDS (row-major, stride 18). Same-wave LDS is in-order.
        #pragma unroll
        for (int v = 0; v < 8; ++v)
            lw[(v + 8 * hi) * 18 + ln] = S[t][v];

        // A frags of P: sub-k q -> lane reads P[ln][4q + 2*hi + {0,1}]
        v2f aP[4];
        #pragma unroll
        for (int q = 0; q < 4; ++q)
            aP[q] = *(const v2f*)(lw + ln * 18 + 4 * q + 2 * hi);

        // O[dc] += P(16x16) * V(16x64), 4 sub-k x 4 d-chunks
        #pragma unroll
        for (int q = 0; q < 4; ++q) {
            const float* vp = Vb + (size_t)(j0 + 4 * q + 2 * hi) * D_;
            #pragma unroll
            for (int dc = 0; dc < 4; ++dc) {
                v2f bV;                                    // B(k,n)=V[j0+4q+k][16dc+n]
                bV.x = vp[16 * dc + ln];
                bV.y = vp[D_ + 16 * dc + ln];
                O[dc] = __builtin_amdgcn_wmma_f32_16x16x4_f32(
                            false, aP[q], false, bV, (short)0, O[dc], false, false);
            }
        }
    }

    // ---- store O (coalesced along d; small + read back soon -> regular) ----
    #pragma unroll
    for (int dc = 0; dc < 4; ++dc)
        #pragma unroll
        for (int v = 0; v < 8; ++v) {
            const int i = i0 + v + 8 * hi;
            Ob[(size_t)i * D_ + 16 * dc + ln] = O[dc][v];
        }
}

// ---------------------------------------------------------------------------
extern "C" void kernel_launch(void* const* d_in, const int* in_sizes, int n_in,
                              void* d_out, int out_size, void* d_ws, size_t ws_size,
                              hipStream_t stream) {
    const float* Q = (const float*)d_in[0];
    const float* K = (const float*)d_in[1];
    const float* V = (const float*)d_in[2];

    float* out  = (float*)d_out;                                // [B,H,L,D]
    float* attn = out + (size_t)B_ * H_ * L_ * D_;              // [B,H,L,L]

    // 1) zero the full attn tensor (402 MB) with nontemporal b128 stores.
    const long long n4 = ((long long)B_ * H_ * L_ * L_) / 4;    // v4f count
    attn_zero_kernel<<<4096, 256, 0, stream>>>((v4f*)attn, n4);

    // 2) band-limited WMMA attention; overwrites the in-band region.
    const int tiles  = B_ * H_ * (L_ / 16);                     // 3072 row tiles
    const int blocks = tiles / WAVES_PER_BLOCK;                 // 768
    local_attn_kernel<<<blocks, 32 * WAVES_PER_BLOCK, 0, stream>>>(Q, K, V, out, attn);
}